// DiceLoss_54176717472355
// MI455X (gfx1250) — compile-verified
//
#include <hip/hip_runtime.h>

#define NUM_CLASSES 19
#define EPS 1e-5f
#define BLOCK 256
#define NREP 8          // one LDS histogram replica per wave32 (256/32)
#define GRIDN 2048

// ---- gfx1250 async global->LDS path (guarded; fallback = direct b128 loads) ----
#if defined(__AMDGCN__) && __has_builtin(__builtin_amdgcn_global_load_async_to_lds_b128) && __has_builtin(__builtin_amdgcn_s_wait_asynccnt)
#define USE_ASYNC 1
#else
#define USE_ASYNC 0
#endif

#if USE_ASYNC
typedef int v4i_t __attribute__((ext_vector_type(4)));
typedef __attribute__((address_space(1))) v4i_t* gbl_v4i_p;
typedef __attribute__((address_space(3))) v4i_t* lds_v4i_p;
#endif

__global__ void dice_zero_ws(unsigned* ws) {
    int t = threadIdx.x;
    if (t < 3 * NUM_CLASSES) ws[t] = 0u;
}

__global__ __launch_bounds__(BLOCK) void dice_hist_kernel(const int* __restrict__ yp,
                                                          const int* __restrict__ yt,
                                                          unsigned* __restrict__ ws,
                                                          long long n) {
    // Packed per-wave histograms: low32 = count_y, high32 = intersection
    __shared__ unsigned long long hYI[NREP][NUM_CLASSES + 1];
    __shared__ unsigned           hP [NREP][NUM_CLASSES + 5];
#if USE_ASYNC
    __shared__ int4 stY[2][BLOCK];
    __shared__ int4 stP[2][BLOCK];
#endif
    const int tid = threadIdx.x;
    const int rep = tid >> 5;   // wave id (wave32)

    for (int i = tid; i < NREP * (NUM_CLASSES + 1); i += BLOCK)
        ((unsigned long long*)hYI)[i] = 0ull;
    for (int i = tid; i < NREP * (NUM_CLASSES + 5); i += BLOCK)
        ((unsigned*)hP)[i] = 0u;
    __syncthreads();

    const long long n4 = n >> 2;
    const long long gstride = (long long)gridDim.x * BLOCK;
    long long i = (long long)blockIdx.x * BLOCK + tid;

#define DICE_PROC(tv, pv)                                                        \
    do {                                                                         \
        int _t = (tv), _p = (pv);                                                \
        unsigned long long _a = 1ull | ((_t == _p) ? (1ull << 32) : 0ull);       \
        atomicAdd(&hYI[rep][_t], _a);                                            \
        atomicAdd(&hP[rep][_p], 1u);                                             \
    } while (0)

#if USE_ASYNC
    // Double-buffered async global->LDS staging (per-lane 128-bit transfers).
  #define DICE_ISSUE(idx, b)                                                     \
    do {                                                                         \
        __builtin_amdgcn_global_load_async_to_lds_b128(                          \
            (gbl_v4i_p)(v4i_t*)((int4*)yp + (idx)),                              \
            (lds_v4i_p)(v4i_t*)&stP[(b)][tid], 0, 0);                            \
        __builtin_amdgcn_global_load_async_to_lds_b128(                          \
            (gbl_v4i_p)(v4i_t*)((int4*)yt + (idx)),                              \
            (lds_v4i_p)(v4i_t*)&stY[(b)][tid], 0, 0);                            \
    } while (0)

    int buf = 0;
    if (i < n4) DICE_ISSUE(i, 0);
    for (long long it = i; it < n4; it += gstride) {
        long long nx = it + gstride;
        if (nx < n4) {
            DICE_ISSUE(nx, buf ^ 1);
            __builtin_amdgcn_s_wait_asynccnt(2);   // current buffer's 2 loads done
        } else {
            __builtin_amdgcn_s_wait_asynccnt(0);   // drain
        }
        __asm__ __volatile__("" ::: "memory");
        int4 p4 = stP[buf][tid];
        int4 t4 = stY[buf][tid];
        buf ^= 1;
        DICE_PROC(t4.x, p4.x); DICE_PROC(t4.y, p4.y);
        DICE_PROC(t4.z, p4.z); DICE_PROC(t4.w, p4.w);
    }
#else
    for (; i < n4; i += gstride) {
        int4 p4 = ((const int4*)yp)[i];
        int4 t4 = ((const int4*)yt)[i];
        DICE_PROC(t4.x, p4.x); DICE_PROC(t4.y, p4.y);
        DICE_PROC(t4.z, p4.z); DICE_PROC(t4.w, p4.w);
    }
#endif
    __syncthreads();

    // Merge wave replicas, push block totals to global bins.
    if (tid < NUM_CLASSES) {
        unsigned cy = 0, in = 0, cp = 0;
        for (int r = 0; r < NREP; ++r) {
            unsigned long long v = hYI[r][tid];
            cy += (unsigned)v;
            in += (unsigned)(v >> 32);
            cp += hP[r][tid];
        }
        if (cy) atomicAdd(&ws[tid], cy);
        if (cp) atomicAdd(&ws[NUM_CLASSES + tid], cp);
        if (in) atomicAdd(&ws[2 * NUM_CLASSES + tid], in);
    }

    // Tail (n not divisible by 4) — handled by one thread directly on global bins.
    if (blockIdx.x == 0 && tid == 0) {
        for (long long k = (n4 << 2); k < n; ++k) {
            int t = yt[k], p = yp[k];
            atomicAdd(&ws[t], 1u);
            atomicAdd(&ws[NUM_CLASSES + p], 1u);
            if (t == p) atomicAdd(&ws[2 * NUM_CLASSES + t], 1u);
        }
    }
}

__global__ void dice_finalize_kernel(const unsigned* __restrict__ ws, float* __restrict__ out) {
    if (blockIdx.x == 0 && threadIdx.x == 0) {
        float acc = 0.f;
        for (int c = 0; c < NUM_CLASSES; ++c) {
            float cy = (float)ws[c];
            float cp = (float)ws[NUM_CLASSES + c];
            float in = (float)ws[2 * NUM_CLASSES + c];
            float uni = cy + cp - in;
            acc += (2.f * in + EPS) / (uni + EPS);
        }
        out[0] = 1.f - acc / (float)NUM_CLASSES;
    }
}

extern "C" void kernel_launch(void* const* d_in, const int* in_sizes, int n_in,
                              void* d_out, int out_size, void* d_ws, size_t ws_size,
                              hipStream_t stream) {
    const int* yp = (const int*)d_in[0];   // y_pred, int32
    const int* yt = (const int*)d_in[1];   // y,      int32
    unsigned* ws  = (unsigned*)d_ws;       // 57 x u32 bins
    long long n   = (long long)in_sizes[0];

    dice_zero_ws<<<1, 64, 0, stream>>>(ws);

    long long n4 = n >> 2;
    long long maxb = (n4 + BLOCK - 1) / BLOCK;
    int blocks = GRIDN;
    if (maxb < blocks) blocks = (int)(maxb > 0 ? maxb : 1);
    dice_hist_kernel<<<blocks, BLOCK, 0, stream>>>(yp, yt, ws, n);

    dice_finalize_kernel<<<1, 32, 0, stream>>>(ws, (float*)d_out);
}